// HarmonicOscillatorWithInteractionDold_89137751261221
// MI455X (gfx1250) — compile-verified
//
#include <hip/hip_runtime.h>
#include <hip/hip_bf16.h>
#include <math.h>

#define W_WALK 8192
#define NPART  6
#define NDIM   3
#define FEAT   18      // NPART*NDIM
#define HID    512
#define GWALK  8       // walkers per lap-block
#define ROWS   288     // 2 * GWALK * FEAT  (A-stack + C-stack)
#define KCH    128     // K-chunk staged in LDS
#define LDS_STRIDE 132 // 128 + 4 pad -> conflict-free ds_load_b64

typedef float v2f __attribute__((ext_vector_type(2)));
typedef float v8f __attribute__((ext_vector_type(8)));

__device__ __forceinline__ v2f ld2(const float* p) { return *(const v2f*)p; }

// ---------------------------------------------------------------- layer 1
// h1[w][j] = tanh(b1[j] + sum_k x[w][k] * W1[k][j])
__global__ void k_fwd1(const float* __restrict__ x, const float* __restrict__ W1,
                       const float* __restrict__ b1, float* __restrict__ h1) {
  __shared__ float xs[FEAT];
  const int w = blockIdx.x;
  if (threadIdx.x < FEAT) xs[threadIdx.x] = x[w * FEAT + threadIdx.x];
  __syncthreads();
  for (int j = threadIdx.x; j < HID; j += blockDim.x) {
    float s = b1[j];
#pragma unroll
    for (int k = 0; k < FEAT; ++k) s += xs[k] * W1[k * HID + j];
    h1[(size_t)w * HID + j] = tanhf(s);
  }
}

// ---------------------------------------------------------------- W2 -> W2^T
__global__ void k_transpose(const float* __restrict__ in, float* __restrict__ out) {
  __shared__ float tile[32][33];
  const int xi = blockIdx.x * 32 + threadIdx.x;  // n
  const int y0 = blockIdx.y * 32;                // j
  for (int i = threadIdx.y; i < 32; i += 8)
    tile[i][threadIdx.x] = in[(size_t)(y0 + i) * HID + xi];
  __syncthreads();
  const int xo = blockIdx.y * 32 + threadIdx.x;  // j
  const int z0 = blockIdx.x * 32;                // n
  for (int i = threadIdx.y; i < 32; i += 8)
    out[(size_t)(z0 + i) * HID + xo] = tile[threadIdx.x][i];
}

// ---------------------------------------------------------------- layer 2 GEMM (WMMA f32)
// z2 = h1 @ W2 + b2 ; h2 = tanh(z2) ; store w3a = W3*(1-h2^2), w3b = -2*W3*h2*(1-h2^2)
// wave = 2 walker M-tiles x 4 N-tiles: 2 A-loads + 4 B-loads per 8 WMMAs.
__global__ void k_fwd2(const float* __restrict__ h1, const float* __restrict__ w2t,
                       const float* __restrict__ b2, const float* __restrict__ W3,
                       float* __restrict__ w3a, float* __restrict__ w3b) {
  const int wid = blockIdx.x * (blockDim.x >> 5) + (threadIdx.x >> 5);
  const int l = threadIdx.x & 31;
  const int mpair = wid >> 3;        // 0..255  (32 walkers each)
  const int nch   = wid & 7;         // 0..7    (64 cols each)
  const int w0 = mpair * 32, n0 = nch * 64;
  const int lm = l & 15;             // M / N index within tile
  const int k0 = (l >> 4) << 1;      // K pair: lanes 0-15 -> K0,K1 ; 16-31 -> K2,K3
  const int half = l >> 4;

  const float* abase[2];
#pragma unroll
  for (int mti = 0; mti < 2; ++mti)
    abase[mti] = h1 + (size_t)(w0 + mti * 16 + lm) * HID + k0;
  const float* bbase[4];
#pragma unroll
  for (int nt = 0; nt < 4; ++nt)
    bbase[nt] = w2t + (size_t)(n0 + nt * 16 + lm) * HID + k0;

  v8f acc[2][4];
#pragma unroll
  for (int mti = 0; mti < 2; ++mti)
#pragma unroll
    for (int nt = 0; nt < 4; ++nt) acc[mti][nt] = (v8f){0.f,0.f,0.f,0.f,0.f,0.f,0.f,0.f};

  for (int j0 = 0; j0 < HID; j0 += 4) {
    v2f a0 = ld2(abase[0] + j0);
    v2f a1 = ld2(abase[1] + j0);
#pragma unroll
    for (int nt = 0; nt < 4; ++nt) {
      v2f b = ld2(bbase[nt] + j0);
      acc[0][nt] = __builtin_amdgcn_wmma_f32_16x16x4_f32(false, a0, false, b,
                                                         (short)0, acc[0][nt], false, false);
      acc[1][nt] = __builtin_amdgcn_wmma_f32_16x16x4_f32(false, a1, false, b,
                                                         (short)0, acc[1][nt], false, false);
    }
  }
#pragma unroll
  for (int mti = 0; mti < 2; ++mti)
#pragma unroll
    for (int nt = 0; nt < 4; ++nt) {
      const int n = n0 + nt * 16 + lm;
      const float w3v = W3[n];
      const float bb  = b2[n];
#pragma unroll
      for (int r = 0; r < 8; ++r) {         // D layout: vgpr r -> M = r (+8 for lanes 16-31)
        const int w = w0 + mti * 16 + r + half * 8;
        const float h  = tanhf(acc[mti][nt][r] + bb);
        const float hp = 1.f - h * h;
        w3a[(size_t)w * HID + n] = w3v * hp;
        w3b[(size_t)w * HID + n] = -2.f * w3v * h * hp;
      }
    }
}

// ---------------------------------------------------------------- Laplacian GEMM (WMMA f32)
// Block = (walker-group of 8, N-chunk of 64), 9 waves.
// A-stack rows 0..143: W1[k,j]*t1, rows 144..287: W1^2*u1.
// wave wv owns M-tiles {2wv, 2wv+1} x all 4 N-tiles: 2 A + 4 B loads per 8 WMMAs.
__global__ void k_lap(const float* __restrict__ h1, const float* __restrict__ W1,
                      const float* __restrict__ w2t, const float* __restrict__ w3a,
                      const float* __restrict__ w3b, float* __restrict__ gbuf,
                      float* __restrict__ lap) {
  extern __shared__ float ldsA[];
  const int wg  = blockIdx.x >> 3;   // walker group
  const int nch = blockIdx.x & 7;
  const int n0  = nch * 64;
  const int tid = threadIdx.x;
  const int wv = tid >> 5, l = tid & 31, lm = l & 15;
  const int k0 = (l >> 4) << 1, half = l >> 4;
  const int mtb = wv * 2;            // first of the wave's two M-tiles

  const float* bptr[4];
  const float* aptr[2];
  v8f acc[2][4];
#pragma unroll
  for (int nt = 0; nt < 4; ++nt)
    bptr[nt] = w2t + (size_t)(n0 + nt * 16 + lm) * HID + k0;
#pragma unroll
  for (int mti = 0; mti < 2; ++mti) {
    aptr[mti] = ldsA + ((mtb + mti) * 16 + lm) * LDS_STRIDE + k0;
#pragma unroll
    for (int nt = 0; nt < 4; ++nt) acc[mti][nt] = (v8f){0.f,0.f,0.f,0.f,0.f,0.f,0.f,0.f};
  }

  for (int kc = 0; kc < 4; ++kc) {
    __syncthreads();  // previous chunk fully consumed
    // generate A-stack chunk into LDS (coalesced in j); 36864/288 = 128 per thread
    for (int idx = tid; idx < ROWS * KCH; idx += 288) {
      const int r = idx >> 7, jj = idx & 127;
      const int j = kc * KCH + jj;
      const int t = (r >= 144);
      const int r2 = r - t * 144;
      const int wl = r2 / FEAT, k = r2 - wl * FEAT;
      const float hv  = h1[(size_t)(wg * GWALK + wl) * HID + j];
      const float tt  = 1.f - hv * hv;            // t1
      const float w1v = W1[k * HID + j];
      ldsA[r * LDS_STRIDE + jj] = t ? (w1v * w1v) * (-2.f * hv * tt) : (w1v * tt);
    }
    __syncthreads();  // chunk visible
    for (int js = 0; js < 32; ++js) {
      const int j0 = js * 4;
      v2f a0 = ld2(aptr[0] + j0);
      v2f a1 = ld2(aptr[1] + j0);
#pragma unroll
      for (int nt = 0; nt < 4; ++nt) {
        v2f b = ld2(bptr[nt] + kc * KCH + j0);
        acc[0][nt] = __builtin_amdgcn_wmma_f32_16x16x4_f32(false, a0, false, b,
                                                           (short)0, acc[0][nt], false, false);
        acc[1][nt] = __builtin_amdgcn_wmma_f32_16x16x4_f32(false, a1, false, b,
                                                           (short)0, acc[1][nt], false, false);
      }
    }
  }

  // epilogue: fold B/S tiles into g[w][k] and lap[w]
#pragma unroll
  for (int mti = 0; mti < 2; ++mti) {
    const int mt = mtb + mti;
    const int t = (mt >= 9);                       // tile is in C-stack (S result)
#pragma unroll
    for (int nt = 0; nt < 4; ++nt) {
      const int n = n0 + nt * 16 + lm;
#pragma unroll
      for (int r = 0; r < 8; ++r) {
        const int row = mt * 16 + r + half * 8;
        const int r2 = row - t * 144;
        const int wl = r2 / FEAT, k = r2 - wl * FEAT;
        const int w = wg * GWALK + wl;
        const float a3 = w3a[(size_t)w * HID + n];
        const float val = acc[mti][nt][r];
        if (!t) {
          const float b3v = w3b[(size_t)w * HID + n];
          float gc = val * a3;          // -> g_k
          float lc = val * val * b3v;   // -> lap (B^2 term)
          for (int off = 1; off < 16; off <<= 1) {
            gc += __shfl_xor(gc, off, 32);
            lc += __shfl_xor(lc, off, 32);
          }
          if (lm == 0) {
            atomicAdd(&gbuf[(size_t)w * FEAT + k], gc);
            atomicAdd(&lap[w], lc);
          }
        } else {
          float lc = val * a3;          // -> lap (S term)
          for (int off = 1; off < 16; off <<= 1) lc += __shfl_xor(lc, off, 32);
          if (lm == 0) atomicAdd(&lap[w], lc);
        }
      }
    }
  }
}

// ---------------------------------------------------------------- finalize
__global__ void k_final(const float* __restrict__ x, const float* __restrict__ gbuf,
                        const float* __restrict__ lap, float* __restrict__ out) {
  const int w = blockIdx.x * blockDim.x + threadIdx.x;
  if (w >= W_WALK) return;
  float gs = 0.f;
#pragma unroll
  for (int k = 0; k < FEAT; ++k) { const float gv = gbuf[(size_t)w * FEAT + k]; gs += gv * gv; }
  const float kin = -0.5f * (lap[w] + gs);
  const float* xw = x + (size_t)w * FEAT;
  float pot = 0.f;
#pragma unroll
  for (int i = 0; i < FEAT; ++i) pot += xw[i] * xw[i];
  pot *= 0.5f;
  float vint = 0.f;
#pragma unroll
  for (int i = 0; i < NPART; ++i)
#pragma unroll
    for (int j = i + 1; j < NPART; ++j) {
      const float d0 = xw[i * 3 + 0] - xw[j * 3 + 0];
      const float d1 = xw[i * 3 + 1] - xw[j * 3 + 1];
      const float d2 = xw[i * 3 + 2] - xw[j * 3 + 2];
      vint += expf(-2.f * (d0 * d0 + d1 * d1 + d2 * d2));  // 1/(2*sigma^2) = 2
    }
  out[w] = kin + pot + 1.5957691216057308f * vint;          // V0/(sqrt(2pi)*sigma)
}

// ---------------------------------------------------------------- launch
extern "C" void kernel_launch(void* const* d_in, const int* in_sizes, int n_in,
                              void* d_out, int out_size, void* d_ws, size_t ws_size,
                              hipStream_t stream) {
  const float* x  = (const float*)d_in[0];
  const float* W1 = (const float*)d_in[1];
  const float* b1 = (const float*)d_in[2];
  const float* W2 = (const float*)d_in[3];
  const float* b2 = (const float*)d_in[4];
  const float* W3 = (const float*)d_in[5];
  // d_in[6] = b3 : only shifts f, not its derivatives -> unused by local energy

  float* ws   = (float*)d_ws;
  float* h1   = ws;                                  // 8192*512
  float* w2t  = h1  + (size_t)W_WALK * HID;          // 512*512
  float* w3a  = w2t + (size_t)HID * HID;             // 8192*512
  float* w3b  = w3a + (size_t)W_WALK * HID;          // 8192*512
  float* gbuf = w3b + (size_t)W_WALK * HID;          // 8192*18
  float* lapb = gbuf + (size_t)W_WALK * FEAT;        // 8192

  k_fwd1<<<W_WALK, 128, 0, stream>>>(x, W1, b1, h1);
  dim3 tb(32, 8), tg(HID / 32, HID / 32);
  k_transpose<<<tg, tb, 0, stream>>>(W2, w2t);
  k_fwd2<<<256, 256, 0, stream>>>(h1, w2t, b2, W3, w3a, w3b);
  hipMemsetAsync(gbuf, 0, (size_t)W_WALK * FEAT * sizeof(float), stream);
  hipMemsetAsync(lapb, 0, (size_t)W_WALK * sizeof(float), stream);
  const size_t ldsBytes = (size_t)ROWS * LDS_STRIDE * sizeof(float);  // ~148.5 KB < 320 KB/WGP
  k_lap<<<(W_WALK / GWALK) * 8, 288, ldsBytes, stream>>>(h1, W1, w2t, w3a, w3b, gbuf, lapb);
  k_final<<<W_WALK / 256, 256, 0, stream>>>(x, gbuf, lapb, (float*)d_out);
}